// _ESM_72387378807110
// MI455X (gfx1250) — compile-verified
//
#include <hip/hip_runtime.h>
#include <hip/hip_bf16.h>
#include <stdint.h>

// ---------------- problem constants (from reference) ----------------
#define BATCH      2048
#define N_TIME     2048
#define INPUT      28
#define OUTPUT     7
#define N_T        1
#define N_S        4
#define PERIOD     7
#define EMBED      9                      // 1 + 1 + PERIOD
#define XLEN       (INPUT + OUTPUT)      // 35 (N_T == 1)
#define FEAT       (INPUT + XLEN + N_S)  // 67
#define WIN_END    (N_TIME - INPUT - OUTPUT + 1)   // 2014
#define NWIN       ((WIN_END + PERIOD - 1) / PERIOD) // 288
#define SEAS_LEN   (PERIOD + 1 + (N_TIME - 1))     // 2055
#define ROWS       64                    // batch rows per insample block

// ---------------- CDNA5 TDM availability ----------------
#if defined(__gfx1250__) && __has_builtin(__builtin_amdgcn_tensor_load_to_lds)
#define HAVE_TDM 1
#else
#define HAVE_TDM 0
#endif

#if HAVE_TDM
typedef unsigned int u32x4 __attribute__((ext_vector_type(4)));
typedef int          i32x8 __attribute__((ext_vector_type(8)));
typedef int          i32x4 __attribute__((ext_vector_type(4)));

__device__ __forceinline__ unsigned lds_byte_addr(const void* p) {
  // Generic pointer to LDS: low 32 bits are the LDS byte offset (flat aperture mapping).
  return (unsigned)(unsigned long long)(uintptr_t)p;
}

// 2D tile load: tile_d0 elements per row, tile_d1 rows, row stride (elements) in the
// global tensor. data_size = 4 bytes. Descriptor packed per CDNA5 ISA D# groups 0/1.
__device__ __forceinline__ void tdm_load_2d(const void* gptr, unsigned ldsb,
                                            unsigned tile_d0, unsigned tile_d1,
                                            unsigned long long stride0) {
  unsigned long long ga = (unsigned long long)(uintptr_t)gptr;
  u32x4 g0;
  g0[0] = 1u;                                         // count=1, user descriptor
  g0[1] = ldsb;                                       // lds_addr
  g0[2] = (unsigned)(ga & 0xFFFFFFFFu);               // global_addr[31:0]
  g0[3] = (unsigned)((ga >> 32) & 0x01FFFFFFu) | 0x80000000u; // addr[56:32] | type=2
  unsigned td0 = tile_d0, td1 = tile_d1;              // tensor dims = tile dims (in-bounds)
  i32x8 g1;
  g1[0] = 0x00020000;                                 // data_size=2 (4 bytes), mask=0
  g1[1] = (int)((td0 & 0xFFFFu) << 16);               // tensor_dim0[15:0]
  g1[2] = (int)((td0 >> 16) | ((td1 & 0xFFFFu) << 16));
  g1[3] = (int)((td1 >> 16) | (tile_d0 << 16));       // tile_dim0
  g1[4] = (int)(tile_d1);                             // tile_dim1, tile_dim2=0
  g1[5] = (int)(unsigned)(stride0 & 0xFFFFFFFFull);   // tensor_dim0_stride[31:0]
  g1[6] = (int)(unsigned)((stride0 >> 32) & 0xFFFFull);
  g1[7] = 0;
  i32x4 z4 = {0, 0, 0, 0};                            // groups 2/3 unused (2D tile)
  i32x8 z8 = {0, 0, 0, 0, 0, 0, 0, 0};                // extra group (clang-23 6-arg form)
  __builtin_amdgcn_tensor_load_to_lds(g0, g1, z4, z4, z8, 0);
}
#endif

// ---------------- kernel 1: ES level/seasonality scan ----------------
__global__ void _ESM_scan(const float* __restrict__ Y,
                          const int* __restrict__ idxs,
                          const float* __restrict__ emb,
                          float* __restrict__ levels,
                          float* __restrict__ seasonal) {
  int b = blockIdx.x * blockDim.x + threadIdx.x;
  if (b >= BATCH) return;
  const float* e = emb + (size_t)idxs[b] * EMBED;
  float lev_sms  = 1.0f / (1.0f + expf(-e[0]));
  float seas_sms = 1.0f / (1.0f + expf(-e[1]));
  float init[PERIOD];
#pragma unroll
  for (int i = 0; i < PERIOD; ++i) init[i] = expf(e[2 + i]);

  const float* Yrow = Y + (size_t)b * N_TIME;
  float* lrow = levels + (size_t)b * N_TIME;
  float* srow = seasonal + (size_t)b * SEAS_LEN;

  float lev = Yrow[0] / init[0];
  float buf[PERIOD];
#pragma unroll
  for (int i = 0; i < PERIOD - 1; ++i) buf[i] = init[i + 1];
  buf[PERIOD - 1] = init[0];

  lrow[0] = lev;
#pragma unroll
  for (int i = 0; i < PERIOD; ++i) srow[i] = init[i];
  srow[PERIOD] = init[0];

  // time loop unrolled by 7 so buf[] indices are compile-time (stays in VGPRs)
#define ESM_STEP(J, K)                                          \
  {                                                             \
    float y = Yrow[(K) + 1];                                    \
    float newlev = lev_sms * (y / buf[J]) + (1.0f - lev_sms) * lev; \
    float news = seas_sms * (y / newlev) + (1.0f - seas_sms) * buf[J]; \
    lev = newlev; buf[J] = news;                                \
    lrow[(K) + 1] = newlev;                                     \
    srow[(PERIOD + 1) + (K)] = news;                            \
  }
  int k = 0;
  for (; k + PERIOD <= N_TIME - 1; k += PERIOD) {
    ESM_STEP(0, k) ESM_STEP(1, k + 1) ESM_STEP(2, k + 2) ESM_STEP(3, k + 3)
    ESM_STEP(4, k + 4) ESM_STEP(5, k + 5) ESM_STEP(6, k + 6)
  }
#pragma unroll
  for (int j = 0; j < PERIOD; ++j) {
    if (k + j < N_TIME - 1) ESM_STEP(j, k + j)
  }
#undef ESM_STEP
}

// ---------------- kernel 2: insample window assembly (TDM-staged) ----------------
__global__ void _ESM_insample(const float* __restrict__ Y,
                              const float* __restrict__ X,
                              const float* __restrict__ S,
                              const float* __restrict__ levels,
                              const float* __restrict__ seasonal,
                              float* __restrict__ out) {
  __shared__ __align__(16) float sY [ROWS * INPUT];
  __shared__ __align__(16) float sSe[ROWS * INPUT];
  __shared__ __align__(16) float sX [ROWS * XLEN];
  __shared__ __align__(16) float sS [ROWS * N_S];
  __shared__ __align__(16) float sL [ROWS];

  const int w  = blockIdx.y;
  const int b0 = blockIdx.x * ROWS;
  const int t0 = w * PERIOD;

#if HAVE_TDM
  if (threadIdx.x < 32) {  // wave 0 only: TDM ignores EXEC, avoid duplicate DMAs
    tdm_load_2d(&Y[(size_t)b0 * N_TIME + t0],        lds_byte_addr(sY),  INPUT, ROWS, N_TIME);
    tdm_load_2d(&seasonal[(size_t)b0 * SEAS_LEN + t0], lds_byte_addr(sSe), INPUT, ROWS, SEAS_LEN);
    tdm_load_2d(&X[(size_t)b0 * N_TIME + t0],        lds_byte_addr(sX),  XLEN,  ROWS, N_TIME);
    tdm_load_2d(&S[(size_t)b0 * N_S],                lds_byte_addr(sS),  ROWS * N_S, 1, ROWS * N_S);
    tdm_load_2d(&levels[(size_t)b0 * N_TIME + t0 + INPUT - 1], lds_byte_addr(sL), 1, ROWS, N_TIME);
#if __has_builtin(__builtin_amdgcn_s_wait_tensorcnt)
    __builtin_amdgcn_s_wait_tensorcnt(0);
#endif
  }
  __syncthreads();
#else
  for (int i = threadIdx.x; i < ROWS * INPUT; i += blockDim.x) {
    int r = i / INPUT, c = i % INPUT;
    sY[i]  = Y[(size_t)(b0 + r) * N_TIME + t0 + c];
    sSe[i] = seasonal[(size_t)(b0 + r) * SEAS_LEN + t0 + c];
  }
  for (int i = threadIdx.x; i < ROWS * XLEN; i += blockDim.x) {
    int r = i / XLEN, c = i % XLEN;
    sX[i] = X[(size_t)(b0 + r) * N_TIME + t0 + c];
  }
  for (int i = threadIdx.x; i < ROWS * N_S; i += blockDim.x)
    sS[i] = S[(size_t)b0 * N_S + i];
  for (int i = threadIdx.x; i < ROWS; i += blockDim.x)
    sL[i] = levels[(size_t)(b0 + i) * N_TIME + t0 + INPUT - 1];
  __syncthreads();
#endif

  // consecutive threads -> consecutive (r,f) -> coalesced stores of (W,B,67)
  for (int i = threadIdx.x; i < ROWS * FEAT; i += blockDim.x) {
    int r = i / FEAT;
    int f = i - r * FEAT;
    float v;
    if (f < INPUT) {
      v = logf(sY[r * INPUT + f] / (sL[r] * sSe[r * INPUT + f]));
    } else if (f < INPUT + XLEN) {
      v = sX[r * XLEN + (f - INPUT)];
    } else {
      v = sS[r * N_S + (f - INPUT - XLEN)];
    }
    out[((size_t)w * BATCH + (b0 + r)) * FEAT + f] = v;
  }
}

// ---------------- kernel 3: outsample gather ----------------
__global__ void _ESM_outsample(const float* __restrict__ Y, float* __restrict__ out) {
  size_t gid = (size_t)blockIdx.x * blockDim.x + threadIdx.x;
  const size_t total = (size_t)NWIN * BATCH * OUTPUT;
  if (gid >= total) return;
  int f = (int)(gid % OUTPUT);
  int b = (int)((gid / OUTPUT) % BATCH);
  int w = (int)(gid / ((size_t)OUTPUT * BATCH));
  out[gid] = Y[(size_t)b * N_TIME + w * PERIOD + INPUT + f];
}

// ---------------- launcher ----------------
extern "C" void kernel_launch(void* const* d_in, const int* in_sizes, int n_in,
                              void* d_out, int out_size, void* d_ws, size_t ws_size,
                              hipStream_t stream) {
  const float* S    = (const float*)d_in[0];
  const float* Y    = (const float*)d_in[1];
  const float* X    = (const float*)d_in[2];
  const int*   idxs = (const int*)  d_in[3];
  const float* emb  = (const float*)d_in[4];
  // d_in[5] = step_size (fixed at 7 by the reference; baked in)

  float* out       = (float*)d_out;
  float* out_ins   = out;                                        // (288,2048,67)
  float* out_outs  = out_ins  + (size_t)NWIN * BATCH * FEAT;     // (288,2048,7)
  float* levels    = out_outs + (size_t)NWIN * BATCH * OUTPUT;   // (2048,2048)
  float* seasonal  = levels   + (size_t)BATCH * N_TIME;          // (2048,2055)

  // 1) sequential ES scan (parallel over batch)
  _ESM_scan<<<dim3(BATCH / 256), dim3(256), 0, stream>>>(Y, idxs, emb, levels, seasonal);

  // 2) insample windows (TDM-staged tiles), one block per (64-row tile, window)
  _ESM_insample<<<dim3(BATCH / ROWS, NWIN), dim3(256), 0, stream>>>(
      Y, X, S, levels, seasonal, out_ins);

  // 3) outsample gather
  {
    size_t total = (size_t)NWIN * BATCH * OUTPUT;
    _ESM_outsample<<<dim3((unsigned)((total + 255) / 256)), dim3(256), 0, stream>>>(Y, out_outs);
  }
}